// HeteroVertexConv_69870527972050
// MI455X (gfx1250) — compile-verified
//
#include <hip/hip_runtime.h>

#define NN 50000
#define NE 800000
#define DD 64
#define NT 4

typedef __attribute__((ext_vector_type(2))) float v2f;
typedef __attribute__((ext_vector_type(8))) float v8f;

// ---------------------------------------------------------------------------
// Kernel 1: zero the output (acts as the aggregation accumulator).
// ---------------------------------------------------------------------------
__global__ void hvc_zero_kernel(float* __restrict__ out, int n) {
    int i = blockIdx.x * blockDim.x + threadIdx.x;
    int idx = i * 4;
    if (idx < n) {
        *(float4*)(out + idx) = make_float4(0.f, 0.f, 0.f, 0.f);
    }
}

// ---------------------------------------------------------------------------
// Kernel 2: edge scatter-add (linearity lets us aggregate BEFORE the weight).
//   out[0      : NN*64 )  <- agg_s  = segment_sum(x[src],   dst)
//   out[NN*64  : NN*256)  <- agg_v  = segment_sum(vec[src], dst)   [N,3,64]
// One thread per (edge, float4-chunk): 64 chunks/edge (16 scalar + 48 vec).
// ---------------------------------------------------------------------------
__global__ void hvc_scatter_kernel(const float* __restrict__ x,
                                   const float* __restrict__ vec,
                                   const int* __restrict__ src,
                                   const int* __restrict__ dst,
                                   float* __restrict__ out) {
    long long w = (long long)blockIdx.x * blockDim.x + threadIdx.x;
    int e = (int)(w >> 6);
    int c = (int)(w & 63);
    if (e >= NE) return;
    int s = src[e];
    int d = dst[e];
    const float* sp;
    float* dp;
    if (c < 16) {
        sp = x + (size_t)s * 64 + c * 4;
        dp = out + (size_t)d * 64 + c * 4;
    } else {
        int cv = c - 16;                      // 0..47 over the [3,64] block
        sp = vec + (size_t)s * 192 + cv * 4;
        dp = out + (size_t)NN * 64 + (size_t)d * 192 + cv * 4;
    }
    float4 v = *(const float4*)sp;
    atomicAdd(dp + 0, v.x);
    atomicAdd(dp + 1, v.y);
    atomicAdd(dp + 2, v.z);
    atomicAdd(dp + 3, v.w);
}

// ---------------------------------------------------------------------------
// Kernel 3: per-node typed projection, in place, using fp32 WMMA.
//   row i of each channel:  out[i] = agg[i] @ W[node_type[i]]
// Block = 16 nodes, 256 threads (8 wave32s). 16 units = 4 channels x 4
// column-chunks; each wave owns 2 units and runs a 4-type x 16-Kstep chain of
// v_wmma_f32_16x16x4_f32 with per-row type masks applied to the A fragment.
// ---------------------------------------------------------------------------
__global__ void hvc_transform_kernel(const int* __restrict__ node_type,
                                     const float* __restrict__ W_s,
                                     const float* __restrict__ W_v,
                                     float* __restrict__ out) {
    __shared__ float A[4 * 16 * 64];   // [channel][row][k]  16 KB
    __shared__ int ntype[16];

    const int tile  = blockIdx.x;
    const int node0 = tile * 16;
    const int tid   = threadIdx.x;

    // Stage the 4 channel tiles (scalar + 3 vector) into LDS; float4 index i
    // maps 1:1 onto A's flat layout (ch*1024 + r*64 + q*4 == i*4).
    for (int i = tid; i < 4 * 16 * 16; i += 256) {
        int ch  = i >> 8;          // / 256
        int rem = i & 255;
        int r   = rem >> 4;
        int q   = rem & 15;
        const float* p;
        if (ch == 0)
            p = out + (size_t)(node0 + r) * 64 + q * 4;
        else
            p = out + (size_t)NN * 64 + (size_t)(node0 + r) * 192 + (ch - 1) * 64 + q * 4;
        ((float4*)A)[i] = *(const float4*)p;
    }
    if (tid < 16) ntype[tid] = node_type[node0 + tid];
    __syncthreads();

    const int wave  = tid >> 5;
    const int lane  = tid & 31;
    const int lhalf = lane >> 4;   // 0: lanes 0-15, 1: lanes 16-31
    const int lmod  = lane & 15;

    for (int u = wave; u < 16; u += 8) {
        const int ch     = u >> 2;       // 0 = scalar, 1..3 = vec channels
        const int nchunk = u & 3;        // 16-column chunk of the output
        const float* Wbase = (ch == 0) ? W_s : W_v;
        const float* Arow  = &A[ch * 1024 + lmod * 64];
        const int    nn    = nchunk * 16 + lmod;      // output column (lane = N)

        v8f acc = {};
        for (int t = 0; t < NT; ++t) {
            const float* Wt = Wbase + (size_t)t * 64 * 64;
            const float  m  = (ntype[lmod] == t) ? 1.0f : 0.0f;  // row-type mask
            for (int k = 0; k < 16; ++k) {
                const int ka = k * 4 + 2 * lhalf;  // ISA A layout: K{0,1}|K{2,3}
                v2f a;
                a.x = Arow[ka]     * m;
                a.y = Arow[ka + 1] * m;
                v2f b;                              // B layout: lane = N, pair = K
                b.x = Wt[(size_t)ka * 64 + nn];
                b.y = Wt[(size_t)(ka + 1) * 64 + nn];
                acc = __builtin_amdgcn_wmma_f32_16x16x4_f32(
                    /*neg_a=*/false, a, /*neg_b=*/false, b,
                    /*c_mod=*/(short)0, acc, /*reuse_a=*/false, /*reuse_b=*/false);
            }
        }

        // D layout: lane = column, VGPR j = row (j + 8*lhalf).
        for (int j = 0; j < 8; ++j) {
            const int M    = j + 8 * lhalf;
            const int node = node0 + M;
            size_t off = (ch == 0)
                ? ((size_t)node * 64 + nn)
                : ((size_t)NN * 64 + (size_t)node * 192 + (ch - 1) * 64 + nn);
            out[off] = acc[j];
        }
    }
}

// ---------------------------------------------------------------------------
extern "C" void kernel_launch(void* const* d_in, const int* in_sizes, int n_in,
                              void* d_out, int out_size, void* d_ws, size_t ws_size,
                              hipStream_t stream) {
    const float* x         = (const float*)d_in[0];
    const float* vec       = (const float*)d_in[1];
    const int*   node_type = (const int*)d_in[2];
    const int*   src       = (const int*)d_in[3];
    const int*   dst       = (const int*)d_in[4];
    const float* W_s       = (const float*)d_in[5];
    const float* W_v       = (const float*)d_in[6];
    float* out = (float*)d_out;

    const int total = NN * 64 + NN * 192;          // 12.8M floats
    hvc_zero_kernel<<<(total / 4 + 255) / 256, 256, 0, stream>>>(out, total);

    const long long work = (long long)NE * 64;     // (edge, chunk) pairs
    hvc_scatter_kernel<<<(int)((work + 255) / 256), 256, 0, stream>>>(
        x, vec, src, dst, out);

    hvc_transform_kernel<<<NN / 16, 256, 0, stream>>>(node_type, W_s, W_v, out);
}